// ActorCriticSpeakerDenseQuantized_38920993636502
// MI455X (gfx1250) — compile-verified
//
#include <hip/hip_runtime.h>
#include <hip/hip_bf16.h>

typedef __attribute__((ext_vector_type(2))) float v2f;
typedef __attribute__((ext_vector_type(8))) float v8f;

// ---------------------------------------------------------------------------
// f32 WMMA 16x16x4 (CDNA5): D = A(16x4) * B(4x16) + C(16x16)
// ---------------------------------------------------------------------------
__device__ __forceinline__ v8f wmma4(v2f a, v2f b, v8f c) {
  return __builtin_amdgcn_wmma_f32_16x16x4_f32(
      /*neg_a=*/false, a, /*neg_b=*/false, b,
      /*c_mod=*/(short)0, c, /*reuse_a=*/false, /*reuse_b=*/false);
}

// MODE: 0 = identity, 1 = relu, 2 = sigmoid, 3 = sigmoid + 1e-8 (SQUISH=1)
template<int MODE>
__device__ __forceinline__ float act(float x) {
  if (MODE == 1) return fmaxf(x, 0.0f);
  if (MODE == 2) return 1.0f / (1.0f + __expf(-x));
  if (MODE == 3) return 1.0f / (1.0f + __expf(-x)) + 1e-8f;
  return x;
}

// Two 16-row C tiles (rows 0-15 and 16-31 of the wave's 32-row slab) sharing
// every B fragment: B-load:WMMA ratio = 1:1 instead of 2:1.
// A: LDS [32 x K] (row stride lda).  B: global row-major [K x N].
template<int K, int N, int MODE>
__device__ __forceinline__ void dense2_lds(const float* A, int lda,
                                           const float* __restrict__ W,
                                           const float* __restrict__ bias,
                                           float* O, int ldo, int lane) {
  const int half = lane >> 4, l16 = lane & 15;
  const float* a0 = A + l16 * lda;          // rows 0-15 tile
  const float* a1 = A + (16 + l16) * lda;   // rows 16-31 tile
  for (int n0 = 0; n0 < N; n0 += 16) {
    const float* bc = W + n0 + l16;
    v8f c0 = {}, c1 = {};
#pragma unroll
    for (int k = 0; k < K; k += 4) {
      const int ka = k + half * 2;
      v2f b;  b.x  = bc[ka * N];  b.y  = bc[(ka + 1) * N];
      v2f x0; x0.x = a0[ka];      x0.y = a0[ka + 1];
      v2f x1; x1.x = a1[ka];      x1.y = a1[ka + 1];
      c0 = wmma4(x0, b, c0);
      c1 = wmma4(x1, b, c1);
    }
    const float bb = bias[n0 + l16];
    float* op0 = O + (half * 8) * ldo + n0 + l16;
    float* op1 = O + (16 + half * 8) * ldo + n0 + l16;
#pragma unroll
    for (int r = 0; r < 8; ++r) {
      op0[r * ldo] = act<MODE>(c0[r] + bb);
      op1[r * ldo] = act<MODE>(c1[r] + bb);
    }
  }
}

template<int K, int N, int MODE>
__device__ __forceinline__ void dense2_glob(const float* A, int lda,
                                            const float* __restrict__ W,
                                            const float* __restrict__ bias,
                                            float* __restrict__ O, long r0, int lane) {
  const int half = lane >> 4, l16 = lane & 15;
  const float* a0 = A + l16 * lda;
  const float* a1 = A + (16 + l16) * lda;
  for (int n0 = 0; n0 < N; n0 += 16) {
    const float* bc = W + n0 + l16;
    v8f c0 = {}, c1 = {};
#pragma unroll
    for (int k = 0; k < K; k += 4) {
      const int ka = k + half * 2;
      v2f b;  b.x  = bc[ka * N];  b.y  = bc[(ka + 1) * N];
      v2f x0; x0.x = a0[ka];      x0.y = a0[ka + 1];
      v2f x1; x1.x = a1[ka];      x1.y = a1[ka + 1];
      c0 = wmma4(x0, b, c0);
      c1 = wmma4(x1, b, c1);
    }
    const float bb = bias[n0 + l16];
    float* op0 = O + (r0 + half * 8) * (long)N + n0 + l16;
    float* op1 = O + (r0 + 16 + half * 8) * (long)N + n0 + l16;
#pragma unroll
    for (int r = 0; r < 8; ++r) {
      op0[(long)r * N] = act<MODE>(c0[r] + bb);
      op1[(long)r * N] = act<MODE>(c1[r] + bb);
    }
  }
}

__global__ void zero_loss_kernel(float* p) {
  if (threadIdx.x == 0 && blockIdx.x == 0) *p = 0.0f;
}

// ---------------------------------------------------------------------------
// Fully fused forward: one wave32 handles 32 batch rows end-to-end.
// LDS: two 32x132 arenas, aliased across phases (z -> arena A, q -> arena B,
// critic hidden layers recycle both once z/q are dead).  ~33 KB/block.
// ---------------------------------------------------------------------------
__global__ __launch_bounds__(32)
void acsdq_fused_kernel(const int* __restrict__ obs, const float* __restrict__ embed,
                        const float* __restrict__ W0, const float* __restrict__ b0,
                        const float* __restrict__ W1, const float* __restrict__ b1,
                        const float* __restrict__ W2, const float* __restrict__ b2,
                        const float* __restrict__ Wp, const float* __restrict__ bp,
                        const float* __restrict__ vq,
                        const float* __restrict__ Wa, const float* __restrict__ ba,
                        const float* __restrict__ Ws, const float* __restrict__ bs,
                        const float* __restrict__ Wc0, const float* __restrict__ bc0,
                        const float* __restrict__ Wc1, const float* __restrict__ bc1,
                        const float* __restrict__ Wc2, const float* __restrict__ bc2,
                        const float* __restrict__ Wc3, const float* __restrict__ bc3,
                        float* __restrict__ out, int Btot) {
  constexpr int LDA = 132;   // 128 + 4 pad: column reads hit distinct LDS banks
  constexpr int LDZ = 68;    // 64 + 4 pad
  constexpr int LDC = 36;    // 32 + 4 pad
  constexpr int ROWS = 32;
  __shared__ float smemA[ROWS * LDA];
  __shared__ float smemB[ROWS * LDA];
  __shared__ int   idxb[ROWS];

  const int  lane = threadIdx.x;
  const int  half = lane >> 4, l16 = lane & 15;
  const long r0   = (long)blockIdx.x * ROWS;

  // phase-aliased views
  float* zbuf  = smemA;   // [32 x LDZ], live: proj .. vq_loss
  float* qbuf  = smemB;   // [32 x LDZ], live: gather .. actor/scale/c0
  float* c2buf = smemA;   // [32 x LDC], live: c2 .. critic dot

  float* actor_o = out;
  float* scale_o = out + (long)Btot * 64;
  float* crit_o  = out + (long)Btot * 128;
  float* loss_o  = out + (long)Btot * 129;
  float* idx_o   = loss_o + 1;

  // ---- class-embedding gather: y = embed[obs] -> smemA [32 x 128] ---------
  for (int t = lane; t < ROWS * 32; t += 32) {       // float4 granules
    const int m = t >> 5, c4 = t & 31;
    const int cls = obs[r0 + m];
    const float4 v = ((const float4*)(embed + (long)cls * 128))[c4];
    *(float4*)(&smemA[m * LDA + c4 * 4]) = v;
  }
  __syncthreads();

  // ---- MLP trunk ----------------------------------------------------------
  dense2_lds<128, 128, 1>(smemA, LDA, W0, b0, smemB, LDA, lane); __syncthreads();
  dense2_lds<128, 128, 1>(smemB, LDA, W1, b1, smemA, LDA, lane); __syncthreads();
  dense2_lds<128, 128, 1>(smemA, LDA, W2, b2, smemB, LDA, lane); __syncthreads();
  dense2_lds<128,  64, 0>(smemB, LDA, Wp, bp, zbuf, LDZ, lane);  __syncthreads();

  // ---- VQ: argmin_n ( ||e_n||^2 - 2 z.e_n )  via WMMA z @ E^T -------------
  float best_s0[8], best_s1[8]; int best_i0[8], best_i1[8];
#pragma unroll
  for (int r = 0; r < 8; ++r) {
    best_s0[r] = INFINITY; best_i0[r] = 0;
    best_s1[r] = INFINITY; best_i1[r] = 0;
  }
  const float* z0 = zbuf + l16 * LDZ;
  const float* z1 = zbuf + (16 + l16) * LDZ;
  for (int n0 = 0; n0 < 512; n0 += 16) {
    const float* vrow = vq + (long)(n0 + l16) * 64;  // B[k][n] = vq[n][k]
    v8f acc0 = {}, acc1 = {};
    float sq = 0.0f;
#pragma unroll
    for (int k = 0; k < 64; k += 4) {
      const int ka = k + half * 2;
      v2f b;  b.x  = vrow[ka]; b.y  = vrow[ka + 1];
      v2f a0; a0.x = z0[ka];   a0.y = z0[ka + 1];
      v2f a1; a1.x = z1[ka];   a1.y = z1[ka + 1];
      sq += b.x * b.x + b.y * b.y;                   // partial ||e_n||^2
      acc0 = wmma4(a0, b, acc0);
      acc1 = wmma4(a1, b, acc1);
    }
    const float e2 = sq + __shfl_xor(sq, 16);        // halves cover complementary K
#pragma unroll
    for (int r = 0; r < 8; ++r) {
      const float s0 = e2 - 2.0f * acc0[r];
      if (s0 < best_s0[r]) { best_s0[r] = s0; best_i0[r] = n0 + l16; }
      const float s1 = e2 - 2.0f * acc1[r];
      if (s1 < best_s1[r]) { best_s1[r] = s1; best_i1[r] = n0 + l16; }
    }
  }
  // cross-lane argmin within each 16-lane half, first-index tie-break
#pragma unroll
  for (int r = 0; r < 8; ++r) {
    float s = best_s0[r]; int i = best_i0[r];
    for (int off = 1; off < 16; off <<= 1) {
      const float os = __shfl_xor(s, off);
      const int   oi = __shfl_xor(i, off);
      if (os < s || (os == s && oi < i)) { s = os; i = oi; }
    }
    if (l16 == 0) idxb[half * 8 + r] = i;
    s = best_s1[r]; i = best_i1[r];
    for (int off = 1; off < 16; off <<= 1) {
      const float os = __shfl_xor(s, off);
      const int   oi = __shfl_xor(i, off);
      if (os < s || (os == s && oi < i)) { s = os; i = oi; }
    }
    if (l16 == 0) idxb[16 + half * 8 + r] = i;
  }
  __syncthreads();

  // ---- gather quantized codes + emit indices ------------------------------
  idx_o[r0 + lane] = (float)idxb[lane];
  for (int t = lane; t < ROWS * 16; t += 32) {       // float4 granules of [32 x 64]
    const int m = t >> 4, c4 = t & 15;
    const float4 v = ((const float4*)(vq + (long)idxb[m] * 64))[c4];
    *(float4*)(&qbuf[m * LDZ + c4 * 4]) = v;
  }
  __syncthreads();

  // ---- vq_loss: q_ste == quantized  =>  1.25 * mean(||z - q||^2) ----------
  float part = 0.0f;
  for (int t = lane; t < ROWS * 64; t += 32) {
    const int m = t >> 6, c = t & 63;
    const float d = zbuf[m * LDZ + c] - qbuf[m * LDZ + c];
    part += d * d;
  }
  for (int off = 16; off > 0; off >>= 1) part += __shfl_xor(part, off);
  if (lane == 0) atomicAdd(loss_o, part * (1.25f / ((float)Btot * 64.0f)));
  __syncthreads();   // zbuf (smemA) dies here; heads may recycle arena A

  // ---- heads (all operate on quantized straight-through value) ------------
  dense2_glob<64, 64, 2>(qbuf, LDZ, Wa, ba, actor_o, r0, lane);
  dense2_glob<64, 64, 3>(qbuf, LDZ, Ws, bs, scale_o, r0, lane);
  dense2_lds <64, 128, 2>(qbuf, LDZ, Wc0, bc0, smemA, LDA, lane); __syncthreads();
  dense2_lds<128, 128, 2>(smemA, LDA, Wc1, bc1, smemB, LDA, lane); __syncthreads();
  dense2_lds<128,  32, 2>(smemB, LDA, Wc2, bc2, c2buf, LDC, lane); __syncthreads();
  {
    float s = bc3[0];
#pragma unroll
    for (int j = 0; j < 32; ++j) s += c2buf[lane * LDC + j] * Wc3[j];
    crit_o[r0 + lane] = s;
  }
}

extern "C" void kernel_launch(void* const* d_in, const int* in_sizes, int n_in,
                              void* d_out, int out_size, void* d_ws, size_t ws_size,
                              hipStream_t stream) {
  const int*   obs   = (const int*)  d_in[0];
  const float* embed = (const float*)d_in[1];
  const float* W0 = (const float*)d_in[2];  const float* b0 = (const float*)d_in[3];
  const float* W1 = (const float*)d_in[4];  const float* b1 = (const float*)d_in[5];
  const float* W2 = (const float*)d_in[6];  const float* b2 = (const float*)d_in[7];
  const float* Wp = (const float*)d_in[8];  const float* bp = (const float*)d_in[9];
  const float* vq = (const float*)d_in[10];
  const float* Wa = (const float*)d_in[11]; const float* ba = (const float*)d_in[12];
  const float* Ws = (const float*)d_in[13]; const float* bs = (const float*)d_in[14];
  const float* Wc0 = (const float*)d_in[15]; const float* bc0 = (const float*)d_in[16];
  const float* Wc1 = (const float*)d_in[17]; const float* bc1 = (const float*)d_in[18];
  const float* Wc2 = (const float*)d_in[19]; const float* bc2 = (const float*)d_in[20];
  const float* Wc3 = (const float*)d_in[21]; const float* bc3 = (const float*)d_in[22];
  float* out = (float*)d_out;
  const int Btot = in_sizes[0];

  zero_loss_kernel<<<1, 32, 0, stream>>>(out + (long)Btot * 129);
  acsdq_fused_kernel<<<Btot / 32, 32, 0, stream>>>(
      obs, embed, W0, b0, W1, b1, W2, b2, Wp, bp, vq,
      Wa, ba, Ws, bs, Wc0, bc0, Wc1, bc1, Wc2, bc2, Wc3, bc3,
      out, Btot);
}